// RGNN_15848429322722
// MI455X (gfx1250) — compile-verified
//
#include <hip/hip_runtime.h>
#include <hip/hip_bf16.h>
#include <math.h>

typedef __attribute__((ext_vector_type(2))) float v2f;
typedef __attribute__((ext_vector_type(8))) float v8f;

#define D_EMB 64
#define C_OUT 32
#define KF    192   // 3 Chebyshev segments * 64 features
#define KP    (KF / 2)          // 96 K-pairs
#define WPAIRS (KP * C_OUT)     // 3072 float2 per weight matrix

// ---------------- utility kernels ----------------
__global__ void k_zero(float* __restrict__ p, long long n) {
    long long i = (long long)blockIdx.x * blockDim.x + threadIdx.x;
    if (i < n) p[i] = 0.f;
}

__global__ void k_neg(const float* __restrict__ src, float* __restrict__ dst, long long n) {
    long long i = (long long)blockIdx.x * blockDim.x + threadIdx.x;
    if (i < n) dst[i] = -src[i];
}

// deg[row[e]] += ew[e]
__global__ void k_deg(const int* __restrict__ row, const float* __restrict__ ew,
                      float* __restrict__ deg, int E) {
    int e = blockIdx.x * blockDim.x + threadIdx.x;
    if (e < E) atomicAdd(&deg[row[e]], ew[e]);
}

__global__ void k_dinv(const float* __restrict__ deg, float* __restrict__ dinv, int N) {
    int i = blockIdx.x * blockDim.x + threadIdx.x;
    if (i < N) { float d = deg[i]; dinv[i] = (d > 0.f) ? rsqrtf(d) : 0.f; }
}

// w_hat[e] = -dinv[row]*ew*dinv[col]
__global__ void k_what(const int* __restrict__ row, const int* __restrict__ col,
                       const float* __restrict__ ew, const float* __restrict__ dinv,
                       float* __restrict__ what, int E) {
    int e = blockIdx.x * blockDim.x + threadIdx.x;
    if (e < E) what[e] = -dinv[row[e]] * ew[e] * dinv[col[e]];
}

// dst[col[e], :] += scale * w_hat[e] * src[row[e], :]   (64 channels, 64 lanes/edge)
__global__ void k_prop(const int* __restrict__ row, const int* __restrict__ col,
                       const float* __restrict__ what, const float* __restrict__ src,
                       float* __restrict__ dst, float scale, int E) {
    int t  = blockIdx.x * 256 + threadIdx.x;
    int e  = t >> 6;          // 4 edges per 256-thread block
    int ch = t & 63;
    if (e < E) {
        float w = what[e] * scale;
        long long r = (long long)row[e] * D_EMB + ch;
        long long c = (long long)col[e] * D_EMB + ch;
        atomicAdd(&dst[c], w * src[r]);
    }
}

// ---------------- fused Chebyshev GEMM + GRU epilogue (WMMA f32) ----------------
// out[n, 0:64] = concat(x,Tx1,Tx2)[n, 0:192] @ [Wz | Wh]   then GRU nonlinearity.
// LDS weight layout is K-paired: lds[p*32 + c] = (W[2p][c], W[2p+1][c]) so every
// B fragment is one aligned ds_load_b64 straight into an even VGPR pair.
__global__ __launch_bounds__(256)
void k_gemm_wmma(const float* __restrict__ x, const float* __restrict__ tx1,
                 const float* __restrict__ tx2,
                 const float* __restrict__ Wz, const float* __restrict__ Wh, // each [192*32], f-major
                 const float* __restrict__ bxz, const float* __restrict__ bhz,
                 const float* __restrict__ bxh, const float* __restrict__ bhh,
                 float* __restrict__ Hout, int nTiles)
{
    __shared__ v2f lds2[2 * WPAIRS];          // 12288 floats = 48 KB
    v2f* ldsZ = lds2;
    v2f* ldsH = lds2 + WPAIRS;
    for (int i = threadIdx.x; i < WPAIRS; i += 256) {
        const int p = i >> 5;                 // K-pair index 0..95
        const int c = i & 31;                 // column
        v2f z; z.x = Wz[(2 * p) * C_OUT + c]; z.y = Wz[(2 * p + 1) * C_OUT + c];
        v2f h; h.x = Wh[(2 * p) * C_OUT + c]; h.y = Wh[(2 * p + 1) * C_OUT + c];
        ldsZ[i] = z;
        ldsH[i] = h;
    }
    __syncthreads();

    const int wave = threadIdx.x >> 5;
    const int lane = threadIdx.x & 31;
    const int tile = blockIdx.x * 8 + wave;
    if (tile >= nTiles) return;               // uniform per-wave: EXEC stays all-ones for WMMA

    const int half = lane >> 4;               // 0: K0/K1, 1: K2/K3
    const int m    = lane & 15;               // A row within tile
    const int nlo  = lane & 15;               // B/D column (low 16)
    const int nhi  = nlo + 16;                // B/D column (high 16)
    const long long rowBase = ((long long)tile * 16 + m) * D_EMB;

    v8f accZlo = {}; v8f accZhi = {}; v8f accHlo = {}; v8f accHhi = {};

    const float* segs[3] = { x, tx1, tx2 };
    #pragma unroll
    for (int s = 0; s < 3; ++s) {
        const float* src = segs[s] + rowBase;
        const int pbase = s * 32;             // K-pair base for this segment (64/2)
        #pragma unroll
        for (int kk = 0; kk < 64; kk += 4) {
            const int k0 = kk + 2 * half;
            v2f a;  a.x = src[k0];  a.y = src[k0 + 1];
            const int prow = (pbase + (kk >> 1) + half) << 5;   // pair*32
            v2f bz0 = ldsZ[prow + nlo];
            v2f bz1 = ldsZ[prow + nhi];
            v2f bh0 = ldsH[prow + nlo];
            v2f bh1 = ldsH[prow + nhi];
            accZlo = __builtin_amdgcn_wmma_f32_16x16x4_f32(false, a, false, bz0, (short)0, accZlo, false, false);
            accZhi = __builtin_amdgcn_wmma_f32_16x16x4_f32(false, a, false, bz1, (short)0, accZhi, false, false);
            accHlo = __builtin_amdgcn_wmma_f32_16x16x4_f32(false, a, false, bh0, (short)0, accHlo, false, false);
            accHhi = __builtin_amdgcn_wmma_f32_16x16x4_f32(false, a, false, bh1, (short)0, accHhi, false, false);
        }
    }

    // GRU epilogue: H = (1 - sigmoid(z)) * tanh(h)  (H0 == 0 identity)
    const float bzlo = bxz[nlo] + bhz[nlo], bzhi = bxz[nhi] + bhz[nhi];
    const float bhlo = bxh[nlo] + bhh[nlo], bhhi = bxh[nhi] + bhh[nhi];
    #pragma unroll
    for (int i = 0; i < 8; ++i) {
        const int node = tile * 16 + i + 8 * half;   // D layout: VGPR i -> M=i (lanes<16) / M=i+8
        float z0 = accZlo[i] + bzlo;
        float h0 = accHlo[i] + bhlo;
        float Z0 = 1.f / (1.f + expf(-z0));
        Hout[(long long)node * C_OUT + nlo] = (1.f - Z0) * tanhf(h0);
        float z1 = accZhi[i] + bzhi;
        float h1 = accHhi[i] + bhhi;
        float Z1 = 1.f / (1.f + expf(-z1));
        Hout[(long long)node * C_OUT + nhi] = (1.f - Z1) * tanhf(h1);
    }
}

// ---------------- softmax over dim 0 of gathered [2B, 32] ----------------
__global__ void k_colsum(const float* __restrict__ H, const int* __restrict__ home,
                         const int* __restrict__ away, float* __restrict__ colsum, int B) {
    __shared__ float red[256];
    const int c = threadIdx.x & 31;
    const int r = threadIdx.x >> 5;          // 0..7
    const int base = blockIdx.x * 64;
    const int totalRows = 2 * B;
    float local = 0.f;
    for (int jj = r; jj < 64; jj += 8) {
        int j = base + jj;
        if (j < totalRows) {
            int node = (j < B) ? home[j] : away[j - B];
            local += expf(H[(long long)node * C_OUT + c]);
        }
    }
    red[threadIdx.x] = local;
    __syncthreads();
    if (r == 0) {
        float s = 0.f;
        #pragma unroll
        for (int q = 0; q < 8; ++q) s += red[q * 32 + c];
        atomicAdd(&colsum[c], s);
    }
}

__global__ void k_out(const float* __restrict__ H, const int* __restrict__ home,
                      const int* __restrict__ away, const float* __restrict__ colsum,
                      float* __restrict__ out, int B) {
    int idx = blockIdx.x * blockDim.x + threadIdx.x;
    int total = 2 * B * C_OUT;
    if (idx >= total) return;
    int j = idx >> 5;
    int c = idx & 31;
    int node = (j < B) ? home[j] : away[j - B];
    out[idx] = expf(H[(long long)node * C_OUT + c]) / colsum[c];
}

// ---------------- launch ----------------
extern "C" void kernel_launch(void* const* d_in, const int* in_sizes, int n_in,
                              void* d_out, int out_size, void* d_ws, size_t ws_size,
                              hipStream_t stream) {
    const int*   edge_index = (const int*)d_in[0];    // [2, E]
    const int*   home       = (const int*)d_in[1];    // [B]
    const int*   away       = (const int*)d_in[2];    // [B]
    const float* ew         = (const float*)d_in[3];  // [E]
    const float* x          = (const float*)d_in[4];  // [N, 64]
    const float* Wxz        = (const float*)d_in[5];  // [3,64,32] -> flat [192,32]
    const float* Wxh        = (const float*)d_in[9];  // [3,64,32] -> flat [192,32]
    const float* bxz        = (const float*)d_in[11];
    const float* bhz        = (const float*)d_in[12];
    const float* bxh        = (const float*)d_in[15];
    const float* bhh        = (const float*)d_in[16];

    const int E = in_sizes[3];
    const int N = in_sizes[4] / D_EMB;
    const int B = in_sizes[1];
    const int* row = edge_index;
    const int* col = edge_index + E;

    // workspace layout (floats): [deg N][Tx1 64N][colsum 32] | dinv N | what E | Tx2 64N | H 32N
    float* ws     = (float*)d_ws;
    float* deg    = ws;
    float* tx1    = ws + N;
    float* colsum = ws + 65LL * N;
    float* dinv   = colsum + 32;
    float* what   = dinv + N;
    float* tx2    = what + E;
    float* Hbuf   = tx2 + 64LL * N;

    const long long zn = 65LL * N + 32;              // deg + Tx1 + colsum contiguous
    k_zero<<<(int)((zn + 255) / 256), 256, 0, stream>>>(ws, zn);

    k_deg <<<(E + 255) / 256, 256, 0, stream>>>(row, ew, deg, E);
    k_dinv<<<(N + 255) / 256, 256, 0, stream>>>(deg, dinv, N);
    k_what<<<(E + 255) / 256, 256, 0, stream>>>(row, col, ew, dinv, what, E);

    // Tx1 = L_hat @ x
    k_prop<<<(E + 3) / 4, 256, 0, stream>>>(row, col, what, x, tx1, 1.0f, E);
    // Tx2 = 2 * L_hat @ Tx1 - x
    const long long nd = 64LL * N;
    k_neg <<<(int)((nd + 255) / 256), 256, 0, stream>>>(x, tx2, nd);
    k_prop<<<(E + 3) / 4, 256, 0, stream>>>(row, col, what, tx1, tx2, 2.0f, E);

    const int nTiles = (N + 15) / 16;
    k_gemm_wmma<<<(nTiles + 7) / 8, 256, 0, stream>>>(x, tx1, tx2, Wxz, Wxh,
                                                      bxz, bhz, bxh, bhh, Hbuf, nTiles);

    k_colsum<<<(2 * B + 63) / 64, 256, 0, stream>>>(Hbuf, home, away, colsum, B);
    const int total = 2 * B * C_OUT;
    k_out<<<(total + 255) / 256, 256, 0, stream>>>(Hbuf, home, away, colsum, (float*)d_out, B);
}